// DSPTeacherChain_3384434229473
// MI455X (gfx1250) — compile-verified
//
#include <hip/hip_runtime.h>
#include <math.h>

#define BATCH   32
#define NSAMP   48000
#define NCHUNK  240
#define CHUNKL  200          // NCHUNK * CHUNKL == NSAMP

typedef __attribute__((ext_vector_type(2))) float v2f;
typedef __attribute__((ext_vector_type(8))) float v8f;

// ---- constants -------------------------------------------------------------
#define K40      0.08304820237218406f    // log2(10)/40   (A = 10^(g/40))
#define L2E2     2.8853900817779268f     // 2*log2(e)     (tanh via exp2)
#define GATEK   -11.541560327111707f     // -8*log2(e)    (sigmoid(8d) via exp2)
#define SQRT2F   1.4142135623730951f
#define INV15    0.06666666666666667f
#define INV_SR   2.0833333333333333e-05f // 1/48000
#define LOW_ARG  (250.0f  * INV_SR)      // w0/(2pi) -> feed raw v_sin/v_cos
#define HIGH_ARG (4000.0f * INV_SR)

// ---- fast scalar helpers (raw CDNA5 transcendentals) -----------------------
__device__ __forceinline__ float frcp (float x){ return __builtin_amdgcn_rcpf(x); }
__device__ __forceinline__ float fex2 (float x){ return __builtin_amdgcn_exp2f(x); }
__device__ __forceinline__ float fsqr (float x){ return __builtin_amdgcn_sqrtf(x); }
__device__ __forceinline__ float fsinr(float x){ return __builtin_amdgcn_sinf(x); } // sin(2*pi*x)
__device__ __forceinline__ float fcosr(float x){ return __builtin_amdgcn_cosf(x); } // cos(2*pi*x)

__device__ __forceinline__ float tanh_fast(float z){
    z = fminf(fmaxf(z, -30.0f), 30.0f);
    float e = fex2(z * L2E2);            // exp(2z)
    return (e - 1.0f) * frcp(e + 1.0f);
}

__device__ __forceinline__ float abs_at(const float* __restrict__ p, int j){
    return (j >= 0 && j < NSAMP) ? fabsf(p[j]) : 0.0f;
}

// ---- per-sample biquad coefficients (RBJ, all three sections) --------------
struct Coefs {
    float b0a,b1a,b2a,a1a,a2a;   // low shelf
    float b0b,b1b,b2b,a1b,a2b;   // mid peak
    float b0c,b1c,b2c,a1c,a2c;   // high shelf
};

__device__ __forceinline__ Coefs make_coefs(float gl, float gm, float fcm, float qm, float gh,
                                            float cwl, float swl, float cwh, float swh){
    Coefs c;
    // low shelf (fixed 250 Hz corner)
    {
        float A = fex2(gl * K40);
        float tt = fsqr(A) * swl * SQRT2F;      // 2*sqrt(A)*sin(w0)*sqrt(2)/2
        float Ap1 = A + 1.0f, Am1 = A - 1.0f;
        float Am1c = Am1 * cwl, Ap1c = Ap1 * cwl;
        float b0 = A * (Ap1 - Am1c + tt);
        float b1 = 2.0f * A * (Am1 - Ap1c);
        float b2 = A * (Ap1 - Am1c - tt);
        float a0 = Ap1 + Am1c + tt;
        float a1 = -2.0f * (Am1 + Ap1c);
        float a2 = Ap1 + Am1c - tt;
        float r = frcp(a0);
        c.b0a = b0*r; c.b1a = b1*r; c.b2a = b2*r; c.a1a = a1*r; c.a2a = a2*r;
    }
    // mid peaking EQ (variable fc, Q)
    {
        float A  = fex2(gm * K40);
        float ag = fcm * INV_SR;
        float cw = fcosr(ag), sw = fsinr(ag);
        float al = sw * 0.5f * frcp(qm);
        float alA = al * A;
        float aoA = al * frcp(A);
        float r = frcp(1.0f + aoA);
        c.b0b = (1.0f + alA) * r;
        c.b1b = (-2.0f * cw) * r;
        c.b2b = (1.0f - alA) * r;
        c.a1b = c.b1b;                           // b1 == a1 for peak EQ
        c.a2b = (1.0f - aoA) * r;
    }
    // high shelf (fixed 4 kHz corner)
    {
        float A = fex2(gh * K40);
        float tt = fsqr(A) * swh * SQRT2F;
        float Ap1 = A + 1.0f, Am1 = A - 1.0f;
        float Am1c = Am1 * cwh, Ap1c = Ap1 * cwh;
        float b0 = A * (Ap1 + Am1c + tt);
        float b1 = -2.0f * A * (Am1 + Ap1c);
        float b2 = A * (Ap1 + Am1c - tt);
        float a0 = Ap1 - Am1c + tt;
        float a1 = 2.0f * (Am1 - Ap1c);
        float a2 = Ap1 - Am1c - tt;
        float r = frcp(a0);
        c.b0c = b0*r; c.b1c = b1*r; c.b2c = b2*r; c.a1c = a1*r; c.a2c = a2*r;
    }
    return c;
}

// one TDF-II step of the 3-section cascade; st[6] = (s1,s2) x 3; returns output
__device__ __forceinline__ float step6(float* st, float v, const Coefs& c){
    float y0 = fmaf(c.b0a, v,  st[0]);
    float n0 = fmaf(c.b1a, v,  st[1]) - c.a1a * y0;
    float n1 = c.b2a * v  - c.a2a * y0;
    float y1 = fmaf(c.b0b, y0, st[2]);
    float n2 = fmaf(c.b1b, y0, st[3]) - c.a1b * y1;
    float n3 = c.b2b * y0 - c.a2b * y1;
    float y2 = fmaf(c.b0c, y1, st[4]);
    float n4 = fmaf(c.b1c, y1, st[5]) - c.a1c * y2;
    float n5 = c.b2c * y1 - c.a2c * y2;
    st[0]=n0; st[1]=n1; st[2]=n2; st[3]=n3; st[4]=n4; st[5]=n5;
    return y2;
}

// ---- Pass 1: per-chunk augmented 7x7 affine transition [Phi p; 0 1] --------
__global__ void __launch_bounds__(256)
k_pass1(const float* __restrict__ x,  const float* __restrict__ gl,
        const float* __restrict__ gm, const float* __restrict__ fcm,
        const float* __restrict__ qm, const float* __restrict__ gh,
        float* __restrict__ ws1){
    int tid = blockIdx.x * blockDim.x + threadIdx.x;
    if (tid >= BATCH * NCHUNK) return;
    int b = tid / NCHUNK, cix = tid % NCHUNK;
    int t0 = cix * CHUNKL;
    const float* xb  = x   + (size_t)b * NSAMP;
    const float* pgl = gl  + (size_t)b * NSAMP;
    const float* pgm = gm  + (size_t)b * NSAMP;
    const float* pfc = fcm + (size_t)b * NSAMP;
    const float* pq  = qm  + (size_t)b * NSAMP;
    const float* pgh = gh  + (size_t)b * NSAMP;

    const float cwl = fcosr(LOW_ARG),  swl = fsinr(LOW_ARG);
    const float cwh = fcosr(HIGH_ARG), swh = fsinr(HIGH_ARG);

    // moving-average running sum for window [t0-7, t0+7]
    float msum = 0.0f;
    #pragma unroll 1
    for (int j = t0 - 7; j <= t0 + 7; ++j) msum += abs_at(xb, j);

    float col[6][6];     // Phi columns (column-major)
    float colp[6];       // affine part p
    #pragma unroll
    for (int j = 0; j < 6; ++j){
        #pragma unroll
        for (int i = 0; i < 6; ++i) col[j][i] = (i == j) ? 1.0f : 0.0f;
        colp[j] = 0.0f;
    }

    #pragma unroll 1
    for (int t = t0; t < t0 + CHUNKL; ++t){
        Coefs cf = make_coefs(pgl[t], pgm[t], pfc[t], pq[t], pgh[t], cwl, swl, cwh, swh);
        float xv = xb[t];
        float xa = fabsf(xv);
        float env = msum * INV15;
        float gate = frcp(1.0f + fex2((xa - env) * GATEK));
        float s = (1.0f - gate) * xv;
        #pragma unroll
        for (int j = 0; j < 6; ++j) (void)step6(col[j], 0.0f, cf);
        (void)step6(colp, s, cf);
        msum += abs_at(xb, t + 8) - abs_at(xb, t - 7);
    }

    float* w = ws1 + (size_t)tid * 56;    // 7x7 row-major, padded stride 56
    #pragma unroll
    for (int r = 0; r < 6; ++r){
        #pragma unroll
        for (int k = 0; k < 6; ++k) w[r*7 + k] = col[k][r];
        w[r*7 + 6] = colp[r];
    }
    #pragma unroll
    for (int k = 0; k < 6; ++k) w[42 + k] = 0.0f;
    w[48] = 1.0f;
}

// ---- Pass 2: prefix product of chunk transitions via f32 WMMA --------------
// One wave per batch row. Running product P lives in LDS (16x16, 7x7 active);
// per chunk: emit x_init = column 6 of P, then P <- PhiTilde_c * P.
// A is zero outside the top-left 7x7, so only K-slices 0..1 (K=0..7) of the
// 16x16x16 product contribute: exactly TWO chained V_WMMA_F32_16X16X4_F32.
__global__ void __launch_bounds__(32)
k_pass2(const float* __restrict__ ws1, float* __restrict__ ws2){
    const int b  = blockIdx.x;
    const int l  = threadIdx.x;
    const int lr = l & 15;
    const bool hi = (l >= 16);
    const int lrc = (lr < 7) ? lr : 6;          // clamped row for branchless loads
    __shared__ float P[256];

    for (int i = l; i < 256; i += 32)
        P[i] = (((i >> 4) == (i & 15)) && ((i >> 4) < 7)) ? 1.0f : 0.0f;
    __syncthreads();

    #pragma unroll 1
    for (int c = 0; c < NCHUNK; ++c){
        // initial state of chunk c = affine column of P_{c-1} (zeros when c==0)
        if (l < 6) ws2[((size_t)b * NCHUNK + c) * 8 + l] = P[l*16 + 6];

        const float* phi = ws1 + ((size_t)b * NCHUNK + c) * 56;
        v2f aS[2], bS[2];
        #pragma unroll
        for (int s4 = 0; s4 < 2; ++s4){
            #pragma unroll
            for (int v = 0; v < 2; ++v){
                int k  = 4*s4 + (hi ? v + 2 : v);
                int kc = (k < 7) ? k : 6;
                // A (16x4 slice): lane=row, vgpr+half=k ; branchless load + select
                float av = phi[lrc*7 + kc];
                aS[s4][v] = (lr < 7 && k < 7) ? av : 0.0f;
                // B (4x16 slice): vgpr+half=k, lane=col
                bS[s4][v] = P[k*16 + lr];
            }
        }
        __syncthreads();

        v8f d = {0.f,0.f,0.f,0.f,0.f,0.f,0.f,0.f};
        d = __builtin_amdgcn_wmma_f32_16x16x4_f32(false, aS[0], false, bS[0], (short)0, d, false, false);
        d = __builtin_amdgcn_wmma_f32_16x16x4_f32(false, aS[1], false, bS[1], (short)0, d, false, false);
        __syncthreads();

        // write D (C/D layout) back into P
        #pragma unroll
        for (int v = 0; v < 8; ++v){
            int row = hi ? v + 8 : v;
            P[row*16 + lr] = d[v];
        }
        __syncthreads();
    }
}

// ---- Pass 3: replay chunks from corrected initial states + elementwise -----
__global__ void __launch_bounds__(256)
k_pass3(const float* __restrict__ x,   const float* __restrict__ alp,
        const float* __restrict__ bet, const float* __restrict__ gl,
        const float* __restrict__ gm,  const float* __restrict__ fcm,
        const float* __restrict__ qm,  const float* __restrict__ gh,
        const float* __restrict__ aa,  const float* __restrict__ ws2,
        float* __restrict__ out){
    int tid = blockIdx.x * blockDim.x + threadIdx.x;
    if (tid >= BATCH * NCHUNK) return;
    int b = tid / NCHUNK, cix = tid % NCHUNK;
    int t0 = cix * CHUNKL;
    const float* xb  = x   + (size_t)b * NSAMP;
    const float* pal = alp + (size_t)b * NSAMP;
    const float* pbe = bet + (size_t)b * NSAMP;
    const float* pgl = gl  + (size_t)b * NSAMP;
    const float* pgm = gm  + (size_t)b * NSAMP;
    const float* pfc = fcm + (size_t)b * NSAMP;
    const float* pq  = qm  + (size_t)b * NSAMP;
    const float* pgh = gh  + (size_t)b * NSAMP;
    const float* paa = aa  + (size_t)b * NSAMP;
    float*       po  = out + (size_t)b * NSAMP;

    const float cwl = fcosr(LOW_ARG),  swl = fsinr(LOW_ARG);
    const float cwh = fcosr(HIGH_ARG), swh = fsinr(HIGH_ARG);

    float st[6];
    {
        const float* si = ws2 + (size_t)tid * 8;
        #pragma unroll
        for (int i = 0; i < 6; ++i) st[i] = si[i];
    }

    float msum = 0.0f;
    #pragma unroll 1
    for (int j = t0 - 7; j <= t0 + 7; ++j) msum += abs_at(xb, j);

    #pragma unroll 1
    for (int t = t0; t < t0 + CHUNKL; ++t){
        Coefs cf = make_coefs(pgl[t], pgm[t], pfc[t], pq[t], pgh[t], cwl, swl, cwh, swh);
        float xv = xb[t];
        float xa = fabsf(xv);
        float env = msum * INV15;
        float gate = frcp(1.0f + fex2((xa - env) * GATEK));
        float at  = gate * xv;
        float s   = (1.0f - gate) * xv;
        float ys  = step6(st, s, cf);
        float bt  = pbe[t];
        float ya  = tanh_fast(fmaf(pal[t], at, bt)) - tanh_fast(bt);
        float av  = paa[t];
        po[t] = av * ya + (1.0f - av) * ys;
        msum += abs_at(xb, t + 8) - abs_at(xb, t - 7);
    }
}

// ---- entry -----------------------------------------------------------------
extern "C" void kernel_launch(void* const* d_in, const int* in_sizes, int n_in,
                              void* d_out, int out_size, void* d_ws, size_t ws_size,
                              hipStream_t stream){
    (void)in_sizes; (void)n_in; (void)out_size; (void)ws_size;
    const float* x   = (const float*)d_in[0];
    const float* alp = (const float*)d_in[1];
    const float* bet = (const float*)d_in[2];
    const float* gl  = (const float*)d_in[3];
    const float* gm  = (const float*)d_in[4];
    const float* fcm = (const float*)d_in[5];
    const float* qm  = (const float*)d_in[6];
    const float* gh  = (const float*)d_in[7];
    const float* aa  = (const float*)d_in[8];
    float* out = (float*)d_out;

    float* ws1 = (float*)d_ws;                            // B*C*56 floats (~1.7 MB)
    float* ws2 = ws1 + (size_t)BATCH * NCHUNK * 56;       // B*C*8 floats

    dim3 blk(256);
    dim3 grd((BATCH * NCHUNK + 255) / 256);
    k_pass1<<<grd, blk, 0, stream>>>(x, gl, gm, fcm, qm, gh, ws1);
    k_pass2<<<dim3(BATCH), dim3(32), 0, stream>>>(ws1, ws2);
    k_pass3<<<grd, blk, 0, stream>>>(x, alp, bet, gl, gm, fcm, qm, gh, aa, ws2, out);
}